// VectorQuantizer_34325378629876
// MI455X (gfx1250) — compile-verified
//
#include <hip/hip_runtime.h>
#include <hip/hip_bf16.h>

typedef __attribute__((ext_vector_type(2))) float v2f;
typedef __attribute__((ext_vector_type(8))) float v8f;
typedef int v4i_g __attribute__((vector_size(4 * sizeof(int))));
typedef __attribute__((address_space(1))) v4i_g* gptr_v4i;
typedef __attribute__((address_space(3))) v4i_g* lptr_v4i;

#define NUM_CODES 8192
#define CODE_DIM  256
#define NROWS     16384          // 8 * 2048
#define TILE_K    16             // codes per LDS tile
#define PAD_STRIDE 260           // 260 % 64 = 4 -> conflict-free B reads; 260*4 % 16 == 0
#define COMMIT_SCALE 1.25f       // codebook_loss + 0.25 * commitment_loss

#if defined(__AMDGCN__) && __has_builtin(__builtin_amdgcn_global_load_async_to_lds_b128)
#define VQ_ASYNC_LDS 1
#else
#define VQ_ASYNC_LDS 0
#endif

// ---------------------------------------------------------------------------
// Kernel 0: codebook squared norms
// ---------------------------------------------------------------------------
__global__ void vq_cnorm_kernel(const float* __restrict__ cb,
                                float* __restrict__ cnorm) {
    int r = blockIdx.x * blockDim.x + threadIdx.x;
    if (r >= NUM_CODES) return;
    const float4* p = (const float4*)(cb + (size_t)r * CODE_DIM);
    float s = 0.f;
#pragma unroll 8
    for (int i = 0; i < CODE_DIM / 4; ++i) {
        float4 v = p[i];
        s += v.x * v.x + v.y * v.y + v.z * v.z + v.w * v.w;
    }
    cnorm[r] = s;
}

// ---------------------------------------------------------------------------
// Kernel 1: fused fp32-WMMA GEMM + running argmin over codes.
// Each wave owns 16 rows of z (A fragments resident in 128 VGPRs) and sweeps
// all 8192 codes in 16-code tiles. Tiles are staged into LDS with
// double-buffered GLOBAL_LOAD_ASYNC_TO_LDS (ASYNCcnt) when available.
// ---------------------------------------------------------------------------
__launch_bounds__(256, 1)
__global__ void vq_argmin_kernel(const float* __restrict__ z,
                                 const float* __restrict__ cb,
                                 const float* __restrict__ cnorm,
                                 int* __restrict__ outIdx) {
#if VQ_ASYNC_LDS
    __shared__ float tile[2][TILE_K * PAD_STRIDE];
#else
    __shared__ float tile[1][TILE_K * PAD_STRIDE];
#endif

    const int tid  = threadIdx.x;
    const int lane = tid & 31;
    const int wave = tid >> 5;
    const int half = lane >> 4;      // 0: lanes 0-15, 1: lanes 16-31
    const int l16  = lane & 15;
    const int rowBase = blockIdx.x * 128 + wave * 16;

    // staging role of this thread: code row r (0..15), dim start d0 (0..240)
    const int r  = tid >> 4;
    const int d0 = (tid & 15) << 4;

    // ---- load A fragments for V_WMMA_F32_16X16X4_F32 ----
    // A is 16x4 per wmma: lanes 0-15 hold (k0,k1) of row l16, lanes 16-31 hold (k2,k3).
    v2f a[CODE_DIM / 4];
    {
        const float* zrow = z + (size_t)(rowBase + l16) * CODE_DIM + half * 2;
#pragma unroll
        for (int q = 0; q < CODE_DIM / 4; ++q)
            a[q] = *(const v2f*)(zrow + q * 4);
    }

    float runMin[8];
    int   runIdx[8];
#pragma unroll
    for (int j = 0; j < 8; ++j) { runMin[j] = __builtin_inff(); runIdx[j] = 0; }

#if VQ_ASYNC_LDS
    // ---- prologue: async-stage chunk 0 into buffer 0 ----
    {
        const float* gsrc = cb + (size_t)r * CODE_DIM + d0;
        float* ldst = &tile[0][r * PAD_STRIDE + d0];
#pragma unroll
        for (int i = 0; i < 4; ++i)
            __builtin_amdgcn_global_load_async_to_lds_b128(
                (gptr_v4i)(gsrc + 4 * i), (lptr_v4i)(ldst + 4 * i), 0, 0);
    }

    int buf = 0;
    for (int c0 = 0; c0 < NUM_CODES; c0 += TILE_K, buf ^= 1) {
        // current chunk's async loads (ours) done -> LDS written; barrier makes
        // every wave's staging visible, and certifies the other buffer is idle.
        asm volatile("s_wait_asynccnt 0x0" ::: "memory");
        __syncthreads();

        // kick off next tile into the other buffer; overlaps with the 64 WMMAs
        if (c0 + TILE_K < NUM_CODES) {
            const float* gsrc = cb + (size_t)(c0 + TILE_K + r) * CODE_DIM + d0;
            float* ldst = &tile[buf ^ 1][r * PAD_STRIDE + d0];
#pragma unroll
            for (int i = 0; i < 4; ++i)
                __builtin_amdgcn_global_load_async_to_lds_b128(
                    (gptr_v4i)(gsrc + 4 * i), (lptr_v4i)(ldst + 4 * i), 0, 0);
        }

        // ---- 16x16 C tile = z[16 rows] . cb[16 codes]^T over D=256 ----
        v8f acc = {};
        const float* brow = &tile[buf][l16 * PAD_STRIDE + half * 2];
#pragma unroll
        for (int q = 0; q < CODE_DIM / 4; ++q) {
            v2f b = *(const v2f*)(brow + q * 4);
            acc = __builtin_amdgcn_wmma_f32_16x16x4_f32(
                false, a[q], false, b, (short)0, acc, false, false);
        }

        // ---- running argmin update: dist' = |c|^2 - 2 * (z.c) ----
        const float cn   = cnorm[c0 + l16];
        const int   cand = c0 + l16;
#pragma unroll
        for (int j = 0; j < 8; ++j) {
            float d = __builtin_fmaf(-2.f, acc[j], cn);
            if (d < runMin[j]) { runMin[j] = d; runIdx[j] = cand; }
        }
    }
#else
    // ---- fallback: synchronous staging (two barriers per tile) ----
    for (int c0 = 0; c0 < NUM_CODES; c0 += TILE_K) {
        __syncthreads();
        {
            const float4* src = (const float4*)(cb + (size_t)(c0 + r) * CODE_DIM + d0);
            float4* dst = (float4*)(&tile[0][r * PAD_STRIDE + d0]);
#pragma unroll
            for (int i = 0; i < 4; ++i) dst[i] = src[i];
        }
        __syncthreads();

        if (c0 + TILE_K < NUM_CODES)
            __builtin_prefetch(cb + (size_t)(c0 + TILE_K + r) * CODE_DIM + d0, 0, 3);

        v8f acc = {};
        const float* brow = &tile[0][l16 * PAD_STRIDE + half * 2];
#pragma unroll
        for (int q = 0; q < CODE_DIM / 4; ++q) {
            v2f b = *(const v2f*)(brow + q * 4);
            acc = __builtin_amdgcn_wmma_f32_16x16x4_f32(
                false, a[q], false, b, (short)0, acc, false, false);
        }

        const float cn   = cnorm[c0 + l16];
        const int   cand = c0 + l16;
#pragma unroll
        for (int j = 0; j < 8; ++j) {
            float d = __builtin_fmaf(-2.f, acc[j], cn);
            if (d < runMin[j]) { runMin[j] = d; runIdx[j] = cand; }
        }
    }
#endif

    // ---- collapse the 16 column-slots per row (xor-shuffle within 16 lanes) ----
#pragma unroll
    for (int j = 0; j < 8; ++j) {
        float m  = runMin[j];
        int   ix = runIdx[j];
#pragma unroll
        for (int s = 8; s >= 1; s >>= 1) {
            float om = __shfl_xor(m, s, 16);
            int   oi = __shfl_xor(ix, s, 16);
            if (om < m || (om == m && oi < ix)) { m = om; ix = oi; }  // jnp.argmin: first occurrence
        }
        if (l16 == 0)
            outIdx[rowBase + half * 8 + j] = ix;   // C VGPR j holds rows j (lanes 0-15) and j+8
    }
}

// ---------------------------------------------------------------------------
// Kernel 2: gather codebook rows + accumulate sum((z - q)^2), emit indices
// ---------------------------------------------------------------------------
__global__ void vq_gather_kernel(const float* __restrict__ z,
                                 const float* __restrict__ cb,
                                 const int* __restrict__ idx,
                                 float* __restrict__ outQ,
                                 float* __restrict__ outIdxF,
                                 float* __restrict__ lossAcc) {
    __shared__ float red[8];
    const int e   = blockIdx.x * 256 + threadIdx.x;
    const int row = e >> 8;
    const int d   = e & 255;
    const int k   = idx[row];
    const float q  = cb[(size_t)k * CODE_DIM + d];
    const float zv = z[e];
    outQ[e] = q;                       // straight-through forward value == quantized
    if (d == 0) outIdxF[row] = (float)k;

    float diff = zv - q;
    float s = diff * diff;
#pragma unroll
    for (int off = 16; off >= 1; off >>= 1)
        s += __shfl_xor(s, off, 32);
    const int lane = threadIdx.x & 31, wv = threadIdx.x >> 5;
    if (lane == 0) red[wv] = s;
    __syncthreads();
    if (threadIdx.x == 0) {
        float t = 0.f;
#pragma unroll
        for (int i = 0; i < 8; ++i) t += red[i];
        atomicAdd(lossAcc, t);
    }
}

__global__ void vq_finalize_kernel(const float* __restrict__ lossAcc,
                                   float* __restrict__ outLoss) {
    *outLoss = COMMIT_SCALE * (*lossAcc) / (float)((size_t)NROWS * CODE_DIM);
}

// ---------------------------------------------------------------------------
extern "C" void kernel_launch(void* const* d_in, const int* in_sizes, int n_in,
                              void* d_out, int out_size, void* d_ws, size_t ws_size,
                              hipStream_t stream) {
    const float* z  = (const float*)d_in[0];   // [8,2048,256]
    const float* cb = (const float*)d_in[1];   // [8192,256]

    float* out      = (float*)d_out;
    float* outQ     = out;                            // 4,194,304 floats
    float* outLoss  = out + (size_t)NROWS * CODE_DIM; // 1 float
    float* outIdxF  = outLoss + 1;                    // 16,384 floats

    float* ws       = (float*)d_ws;
    float* lossAcc  = ws;                             // ws[0]
    float* cnorm    = ws + 16;                        // 8192 floats
    int*   idx      = (int*)(ws + 16 + NUM_CODES);    // 16384 ints

    (void)hipMemsetAsync(lossAcc, 0, sizeof(float), stream);

    vq_cnorm_kernel<<<NUM_CODES / 256, 256, 0, stream>>>(cb, cnorm);
    vq_argmin_kernel<<<NROWS / 128, 256, 0, stream>>>(z, cb, cnorm, idx);
    vq_gather_kernel<<<(NROWS * CODE_DIM) / 256, 256, 0, stream>>>(
        z, cb, idx, outQ, outIdxF, lossAcc);
    vq_finalize_kernel<<<1, 1, 0, stream>>>(lossAcc, outLoss);
}